// Interpolator2D_78769700209152
// MI455X (gfx1250) — compile-verified
//
#include <hip/hip_runtime.h>

#define NXg 512
#define NYg 512
#define NCc 8
#define NQq 524288

typedef float v2f __attribute__((ext_vector_type(2)));
typedef float v8f __attribute__((ext_vector_type(8)));

// Lane-major WMMA A-operand table for W = A_BICUBIC^T * T.
// AL[lane][2t+v] = A_BICUBIC[4t + 2*(lane>=16) + v][lane%16]
// (A-matrix 16x4 f32 layout: VGPR v holds K = 4t + 2*(lane>=16) + v, M = lane%16)
__device__ __constant__ float AL[32][8] = {
  /* lane  0 (q= 0) */ { 1, 0, 0, 0, -3, 0,  2, 0},
  /* lane  1 (q= 1) */ { 0, 0, 0, 0,  0, 0,  0, 0},
  /* lane  2 (q= 2) */ { 0, 0, 0, 0,  3, 0, -2, 0},
  /* lane  3 (q= 3) */ { 0, 0, 0, 0,  0, 0,  0, 0},
  /* lane  4 (q= 4) */ { 0, 1, 0, 0,  0,-3,  0, 2},
  /* lane  5 (q= 5) */ { 0, 0, 0, 0,  0, 0,  0, 0},
  /* lane  6 (q= 6) */ { 0, 0, 0, 0,  0, 3,  0,-2},
  /* lane  7 (q= 7) */ { 0, 0, 0, 0,  0, 0,  0, 0},
  /* lane  8 (q= 8) */ { 0, 0, 1, 0, -2, 0,  1, 0},
  /* lane  9 (q= 9) */ { 0, 0, 0, 0,  0, 0,  0, 0},
  /* lane 10 (q=10) */ { 0, 0, 0, 0, -1, 0,  1, 0},
  /* lane 11 (q=11) */ { 0, 0, 0, 0,  0, 0,  0, 0},
  /* lane 12 (q=12) */ { 0, 0, 0, 1,  0,-2,  0, 1},
  /* lane 13 (q=13) */ { 0, 0, 0, 0,  0, 0,  0, 0},
  /* lane 14 (q=14) */ { 0, 0, 0, 0,  0,-1,  0, 1},
  /* lane 15 (q=15) */ { 0, 0, 0, 0,  0, 0,  0, 0},
  /* lane 16 (q= 0) */ {-3, 2, 0, 0,  9,-6, -6, 4},
  /* lane 17 (q= 1) */ { 3,-2, 0, 0, -9, 6,  6,-4},
  /* lane 18 (q= 2) */ { 0, 0, 0, 0, -9, 6,  6,-4},
  /* lane 19 (q= 3) */ { 0, 0, 0, 0,  9,-6, -6, 4},
  /* lane 20 (q= 4) */ {-2, 1, 0, 0,  6,-3, -4, 2},
  /* lane 21 (q= 5) */ {-1, 1, 0, 0,  3,-3, -2, 2},
  /* lane 22 (q= 6) */ { 0, 0, 0, 0, -6, 3,  4,-2},
  /* lane 23 (q= 7) */ { 0, 0, 0, 0, -3, 3,  2,-2},
  /* lane 24 (q= 8) */ { 0, 0,-3, 2,  6,-4, -3, 2},
  /* lane 25 (q= 9) */ { 0, 0, 3,-2, -6, 4,  3,-2},
  /* lane 26 (q=10) */ { 0, 0, 0, 0,  3,-2, -3, 2},
  /* lane 27 (q=11) */ { 0, 0, 0, 0, -3, 2,  3,-2},
  /* lane 28 (q=12) */ { 0, 0,-2, 1,  4,-2, -2, 1},
  /* lane 29 (q=13) */ { 0, 0,-1, 1,  2,-2, -1, 1},
  /* lane 30 (q=14) */ { 0, 0, 0, 0,  2,-1, -2, 1},
  /* lane 31 (q=15) */ { 0, 0, 0, 0,  1,-1, -1, 1}
};

struct F8 { float v[NCc]; };

__device__ inline F8 ld8(const float* p) {
  F8 r;
  const float4 a = *reinterpret_cast<const float4*>(p);
  const float4 b = *reinterpret_cast<const float4*>(p + 4);
  r.v[0]=a.x; r.v[1]=a.y; r.v[2]=a.z; r.v[3]=a.w;
  r.v[4]=b.x; r.v[5]=b.y; r.v[6]=b.z; r.v[7]=b.w;
  return r;
}

__device__ inline void st8(float* p, const F8& r) {
  *reinterpret_cast<float4*>(p)     = make_float4(r.v[0], r.v[1], r.v[2], r.v[3]);
  *reinterpret_cast<float4*>(p + 4) = make_float4(r.v[4], r.v[5], r.v[6], r.v[7]);
}

// ---------------------------------------------------------------------------
// Kernel 1: compute fx, fy, fxy and pack per grid point:
//   G[(ix*NY+iy)*32 + {0..7}=f, {8..15}=fx, {16..23}=fy, {24..31}=fxy]
// One thread per grid point (all 8 channels). 128B per point = 1 cache line.
// ---------------------------------------------------------------------------
__global__ __launch_bounds__(256)
void deriv_pack_kernel(const float* __restrict__ f,
                       const float* __restrict__ x,
                       const float* __restrict__ y,
                       float* __restrict__ G)
{
  const int id = blockIdx.x * blockDim.x + threadIdx.x;
  if (id >= NXg * NYg) return;
  const int ix = id / NYg;
  const int iy = id % NYg;

  auto dinv = [](float a, float b) { float d = b - a; return (d == 0.f) ? 0.f : 1.f / d; };
  auto fv = [&](int px, int py) { return ld8(f + (size_t)(px * NYg + py) * NCc); };

  // y-slope estimate at (px, iy)  (interpax _approx_df, axis=1)
  auto fy_at = [&](int px) {
    F8 s;
    if (iy == 0) {
      F8 a = fv(px, 0), b = fv(px, 1);
      const float w = dinv(y[0], y[1]);
      #pragma unroll
      for (int c = 0; c < NCc; ++c) s.v[c] = (b.v[c] - a.v[c]) * w;
    } else if (iy == NYg - 1) {
      F8 a = fv(px, NYg - 2), b = fv(px, NYg - 1);
      const float w = dinv(y[NYg - 2], y[NYg - 1]);
      #pragma unroll
      for (int c = 0; c < NCc; ++c) s.v[c] = (b.v[c] - a.v[c]) * w;
    } else {
      F8 a = fv(px, iy - 1), b = fv(px, iy), d = fv(px, iy + 1);
      const float w0 = dinv(y[iy - 1], y[iy]), w1 = dinv(y[iy], y[iy + 1]);
      #pragma unroll
      for (int c = 0; c < NCc; ++c)
        s.v[c] = 0.5f * ((b.v[c] - a.v[c]) * w0 + (d.v[c] - b.v[c]) * w1);
    }
    return s;
  };

  // fx at (ix, iy)  (axis=0)
  F8 fxv;
  if (ix == 0) {
    F8 a = fv(0, iy), b = fv(1, iy);
    const float w = dinv(x[0], x[1]);
    #pragma unroll
    for (int c = 0; c < NCc; ++c) fxv.v[c] = (b.v[c] - a.v[c]) * w;
  } else if (ix == NXg - 1) {
    F8 a = fv(NXg - 2, iy), b = fv(NXg - 1, iy);
    const float w = dinv(x[NXg - 2], x[NXg - 1]);
    #pragma unroll
    for (int c = 0; c < NCc; ++c) fxv.v[c] = (b.v[c] - a.v[c]) * w;
  } else {
    F8 a = fv(ix - 1, iy), b = fv(ix, iy), d = fv(ix + 1, iy);
    const float w0 = dinv(x[ix - 1], x[ix]), w1 = dinv(x[ix], x[ix + 1]);
    #pragma unroll
    for (int c = 0; c < NCc; ++c)
      fxv.v[c] = 0.5f * ((b.v[c] - a.v[c]) * w0 + (d.v[c] - b.v[c]) * w1);
  }

  F8 fyv = fy_at(ix);

  // fxy = _approx_df(x, fy, axis=0)
  F8 fxyv;
  if (ix == 0) {
    F8 a = fyv, b = fy_at(1);
    const float w = dinv(x[0], x[1]);
    #pragma unroll
    for (int c = 0; c < NCc; ++c) fxyv.v[c] = (b.v[c] - a.v[c]) * w;
  } else if (ix == NXg - 1) {
    F8 a = fy_at(NXg - 2), b = fyv;
    const float w = dinv(x[NXg - 2], x[NXg - 1]);
    #pragma unroll
    for (int c = 0; c < NCc; ++c) fxyv.v[c] = (b.v[c] - a.v[c]) * w;
  } else {
    F8 a = fy_at(ix - 1), b = fyv, d = fy_at(ix + 1);
    const float w0 = dinv(x[ix - 1], x[ix]), w1 = dinv(x[ix], x[ix + 1]);
    #pragma unroll
    for (int c = 0; c < NCc; ++c)
      fxyv.v[c] = 0.5f * ((b.v[c] - a.v[c]) * w0 + (d.v[c] - b.v[c]) * w1);
  }

  float* g = G + (size_t)id * 32;
  st8(g +  0, fv(ix, iy));
  st8(g +  8, fxv);
  st8(g + 16, fyv);
  st8(g + 24, fxyv);
}

// ---------------------------------------------------------------------------
// Kernel 2: WMMA-weighted bicubic gather.
// Each wave32 handles 16 queries: lane pair (n, n+16) shares query n.
// Weights W[q][n] = sum_p A_BICUBIC[p][q] * T_n[p] via 4 chained
// v_wmma_f32_16x16x4_f32 (K=16). D layout: lane n -> weights 0-7 (f, fx),
// lane n+16 -> weights 8-15 (fy, fxy). Each lane gathers its contiguous
// 64B half of the packed 128B corner line. Branch-free operand setup.
// ---------------------------------------------------------------------------
__global__ __launch_bounds__(256)
void bicubic_eval_kernel(const float* __restrict__ xq,
                         const float* __restrict__ yq,
                         const float* __restrict__ x,
                         const float* __restrict__ y,
                         const float* __restrict__ G,
                         float* __restrict__ out)
{
  __shared__ float sx[NXg];
  __shared__ float sy[NYg];
  for (int t = threadIdx.x; t < NXg; t += blockDim.x) sx[t] = x[t];
  for (int t = threadIdx.x; t < NYg; t += blockDim.x) sy[t] = y[t];
  __syncthreads();

  const int tid  = blockIdx.x * blockDim.x + threadIdx.x;
  const int lane = threadIdx.x & 31;
  const int half = lane >> 4;              // 0: f/fx weights, 1: fy/fxy weights
  const int qn   = lane & 15;              // query slot within the wave
  const int qi   = (tid >> 5) * 16 + qn;   // exact fit: grid*block/2 == NQq

  // Per-lane constant A-operand: 8 floats = two B128 loads.
  const float4 al0 = *reinterpret_cast<const float4*>(&AL[lane][0]);
  const float4 al1 = *reinterpret_cast<const float4*>(&AL[lane][4]);

  const float xv = xq[qi];
  const float yv = yq[qi];

  // searchsorted(side='right') then clip to [1, N-1]
  int lo = 0, hi = NXg;
  #pragma unroll
  for (int it = 0; it < 9; ++it) { const int mid = (lo + hi) >> 1; if (sx[mid] <= xv) lo = mid + 1; else hi = mid; }
  const int i = min(max(lo, 1), NXg - 1);
  lo = 0; hi = NYg;
  #pragma unroll
  for (int it = 0; it < 9; ++it) { const int mid = (lo + hi) >> 1; if (sy[mid] <= yv) lo = mid + 1; else hi = mid; }
  const int j = min(max(lo, 1), NYg - 1);

  const float dxc = sx[i] - sx[i - 1];
  const float dyc = sy[j] - sy[j - 1];
  const float tx = (xv - sx[i - 1]) * ((dxc == 0.f) ? 0.f : 1.f / dxc);
  const float ty = (yv - sy[j - 1]) * ((dyc == 0.f) ? 0.f : 1.f / dyc);

  // B operand, no T[] array: T[4t+2h+v] = tx^(2h+v) * ty^t.
  const float tx2 = tx * tx;
  const float hx = half ? tx2       : 1.f;   // tx^(2h)
  const float hy = half ? tx2 * tx  : tx;    // tx^(2h+1)
  const float ty2 = ty * ty;
  const float up0 = 1.f, up1 = ty, up2 = ty2, up3 = ty2 * ty;

  v8f w = {0.f, 0.f, 0.f, 0.f, 0.f, 0.f, 0.f, 0.f};
  {
    v2f a, b;
    a.x = al0.x; a.y = al0.y; b.x = hx * up0; b.y = hy * up0;
    w = __builtin_amdgcn_wmma_f32_16x16x4_f32(false, a, false, b, (short)0, w, false, false);
    a.x = al0.z; a.y = al0.w; b.x = hx * up1; b.y = hy * up1;
    w = __builtin_amdgcn_wmma_f32_16x16x4_f32(false, a, false, b, (short)0, w, false, false);
    a.x = al1.x; a.y = al1.y; b.x = hx * up2; b.y = hy * up2;
    w = __builtin_amdgcn_wmma_f32_16x16x4_f32(false, a, false, b, (short)0, w, false, false);
    a.x = al1.z; a.y = al1.w; b.x = hx * up3; b.y = hy * up3;
    w = __builtin_amdgcn_wmma_f32_16x16x4_f32(false, a, false, b, (short)0, w, false, false);
  }

  // Fold derivative scaling into weights (branch-free):
  // half==0: w[0..3] f (*1), w[4..7] fx (*dx); half==1: w[0..3] fy (*dy), w[4..7] fxy (*dx*dy)
  const float s_lo = half ? dyc : 1.f;
  const float s_hi = half ? dxc * dyc : dxc;
  float w8[8];
  w8[0] = w[0] * s_lo; w8[1] = w[1] * s_lo; w8[2] = w[2] * s_lo; w8[3] = w[3] * s_lo;
  w8[4] = w[4] * s_hi; w8[5] = w[5] * s_hi; w8[6] = w[6] * s_hi; w8[7] = w[7] * s_hi;

  // Gather 4 corners; each lane reads its 64B half ([f,fx] or [fy,fxy]) of the
  // 128B packed line and accumulates 8 channels.
  float acc[8] = {0.f, 0.f, 0.f, 0.f, 0.f, 0.f, 0.f, 0.f};
  const float* Gh = G + half * 16;
  #pragma unroll
  for (int jj = 0; jj < 2; ++jj) {
    #pragma unroll
    for (int ii = 0; ii < 2; ++ii) {
      const int px = i - 1 + ii;
      const int py = j - 1 + jj;
      const float* g = Gh + (size_t)(px * NYg + py) * 32;
      const float4 a0 = *reinterpret_cast<const float4*>(g + 0);
      const float4 a1 = *reinterpret_cast<const float4*>(g + 4);
      const float4 b0 = *reinterpret_cast<const float4*>(g + 8);
      const float4 b1 = *reinterpret_cast<const float4*>(g + 12);
      const float wa = w8[ii + 2 * jj];
      const float wb = w8[4 + ii + 2 * jj];
      acc[0] = fmaf(wa, a0.x, fmaf(wb, b0.x, acc[0]));
      acc[1] = fmaf(wa, a0.y, fmaf(wb, b0.y, acc[1]));
      acc[2] = fmaf(wa, a0.z, fmaf(wb, b0.z, acc[2]));
      acc[3] = fmaf(wa, a0.w, fmaf(wb, b0.w, acc[3]));
      acc[4] = fmaf(wa, a1.x, fmaf(wb, b1.x, acc[4]));
      acc[5] = fmaf(wa, a1.y, fmaf(wb, b1.y, acc[5]));
      acc[6] = fmaf(wa, a1.z, fmaf(wb, b1.z, acc[6]));
      acc[7] = fmaf(wa, a1.w, fmaf(wb, b1.w, acc[7]));
    }
  }

  // Combine lane pair (xor 16); lane<16 stores channels 0-3, lane>=16 stores 4-7.
  float tot[8];
  #pragma unroll
  for (int c = 0; c < 8; ++c) tot[c] = acc[c] + __shfl_xor(acc[c], 16, 32);

  float4 res;
  res.x = half ? tot[4] : tot[0];
  res.y = half ? tot[5] : tot[1];
  res.z = half ? tot[6] : tot[2];
  res.w = half ? tot[7] : tot[3];
  *reinterpret_cast<float4*>(out + (size_t)qi * NCc + half * 4) = res;
}

extern "C" void kernel_launch(void* const* d_in, const int* in_sizes, int n_in,
                              void* d_out, int out_size, void* d_ws, size_t ws_size,
                              hipStream_t stream) {
  (void)in_sizes; (void)n_in; (void)out_size; (void)ws_size;
  const float* xq = (const float*)d_in[0];
  const float* yq = (const float*)d_in[1];
  const float* x  = (const float*)d_in[2];
  const float* y  = (const float*)d_in[3];
  const float* f  = (const float*)d_in[4];
  float* out = (float*)d_out;
  float* G   = (float*)d_ws;   // needs 512*512*32*4 B = 33.5 MB scratch

  deriv_pack_kernel<<<(NXg * NYg + 255) / 256, 256, 0, stream>>>(f, x, y, G);

  // 2 threads per query (lane pair): NQq * 2 / 256 blocks, exact fit.
  bicubic_eval_kernel<<<(NQq * 2) / 256, 256, 0, stream>>>(xq, yq, x, y, G, out);
}